// Quanv_43791486550370
// MI455X (gfx1250) — compile-verified
//
#include <hip/hip_runtime.h>
#include <hip/hip_bf16.h>
#include <stdint.h>

#define NQ   4
#define HW   512
#define OW   511
#define NPIX (OW * OW)

typedef float v2f __attribute__((ext_vector_type(2)));
typedef float v8f __attribute__((ext_vector_type(8)));

// Packed gate list: type[1:0] (0=RX,1=RY,2=RZ,3=CNOT), w0[3:2], w1[5:4], layer[6], widx[8:7]
struct OpList { int n; unsigned int ops[63]; };

// ---------------------------------------------------------------------------
// Setup kernel: fold the whole circuit into one 16x16 complex unitary U.
// Writes Ur (row-major 16x16) at umat[0..255], Ui at umat[256..511].
// ---------------------------------------------------------------------------
__global__ void quanv_build_u(const float* __restrict__ w,
                              float* __restrict__ umat, OpList ol)
{
  if (threadIdx.x != 0 || blockIdx.x != 0) return;
  float ur[16][16], ui[16][16];
  for (int a = 0; a < 16; ++a)
    for (int b = 0; b < 16; ++b) { ur[a][b] = (a == b) ? 1.f : 0.f; ui[a][b] = 0.f; }

  for (int o = 0; o < ol.n; ++o) {
    unsigned e = ol.ops[o];
    int type = e & 3, w0 = (e >> 2) & 3, w1 = (e >> 4) & 3;
    int lay = (e >> 6) & 1, wi = (e >> 7) & 3;
    if (type == 3) {            // CNOT: U <- P*U is a row permutation (swap pairs)
      int cb = 3 - w0, tb = 3 - w1;
      for (int a = 0; a < 16; ++a) {
        if (((a >> cb) & 1) && !((a >> tb) & 1)) {
          int a2 = a | (1 << tb);
          for (int c = 0; c < 16; ++c) {
            float tr = ur[a][c], ti = ui[a][c];
            ur[a][c] = ur[a2][c]; ui[a][c] = ui[a2][c];
            ur[a2][c] = tr;       ui[a2][c] = ti;
          }
        }
      }
    } else {                    // 1-qubit rotation on wire w0
      float th = w[lay * 4 + wi] * 0.5f;
      float ch = cosf(th), sh = sinf(th);
      float g00r, g00i, g01r, g01i, g10r, g10i, g11r, g11i;
      if (type == 0) {          // RX
        g00r = ch; g00i = 0.f; g01r = 0.f; g01i = -sh;
        g10r = 0.f; g10i = -sh; g11r = ch; g11i = 0.f;
      } else if (type == 1) {   // RY
        g00r = ch; g00i = 0.f; g01r = -sh; g01i = 0.f;
        g10r = sh; g10i = 0.f; g11r = ch;  g11i = 0.f;
      } else {                  // RZ
        g00r = ch; g00i = -sh; g01r = 0.f; g01i = 0.f;
        g10r = 0.f; g10i = 0.f; g11r = ch; g11i = sh;
      }
      int bp = 3 - w0;
      for (int a = 0; a < 16; ++a) {
        if (((a >> bp) & 1) == 0) {
          int a1 = a | (1 << bp);
          for (int c = 0; c < 16; ++c) {
            float u0r = ur[a][c], u0i = ui[a][c];
            float u1r = ur[a1][c], u1i = ui[a1][c];
            ur[a][c]  = g00r * u0r - g00i * u0i + g01r * u1r - g01i * u1i;
            ui[a][c]  = g00r * u0i + g00i * u0r + g01r * u1i + g01i * u1r;
            ur[a1][c] = g10r * u0r - g10i * u0i + g11r * u1r - g11i * u1i;
            ui[a1][c] = g10r * u0i + g10i * u0r + g11r * u1i + g11i * u1r;
          }
        }
      }
    }
  }
  for (int a = 0; a < 16; ++a)
    for (int c = 0; c < 16; ++c) {
      umat[a * 16 + c]       = ur[a][c];
      umat[256 + a * 16 + c] = ui[a][c];
    }
}

// lane <-> lane^16 exchange via ds_swizzle (group-of-32 SWAPX16 pattern)
__device__ __forceinline__ float swap16(float v) {
  int r = __builtin_amdgcn_ds_swizzle(__float_as_int(v), 0x401f);
  return __int_as_float(r);
}

// ---------------------------------------------------------------------------
// Main kernel: one wave handles 16 output pixels; per channel: build Psi
// fragments directly in registers (B[kc][v] = t0[kc] * t1[v + 2*half]),
// Yr = Ur*Psi, Yi = Ui*Psi via chained V_WMMA_F32_16X16X4_F32, then probs
// + Z-sign reductions (half-swap via ds_swizzle), summed over channels.
// ---------------------------------------------------------------------------
__global__ __launch_bounds__(256) void quanv_main(const float* __restrict__ x,
                                                  const float* __restrict__ umat,
                                                  float* __restrict__ out)
{
  const int lane = threadIdx.x & 31;
  const int wave = threadIdx.x >> 5;
  const int half = lane >> 4;    // 0: rows M=0..7 / K {0,1} ; 1: rows M=8..15 / K {2,3}
  const int col  = lane & 15;    // B/D column (pixel slot), A row

  long tile = (long)blockIdx.x * 8 + wave;
  int pix   = (int)(tile * 16) + col;
  bool valid = pix < NPIX;
  int pc = valid ? pix : (NPIX - 1);
  int pi = pc / OW;
  int pj = pc - pi * OW;

  // A fragments (Ur, Ui) per ISA 16x4 f32 layout: lane L<16 -> K={0,1}, L>=16 -> K={2,3}
  const int cb = 2 * half;
  v2f Ar[4], Ai[4];
#pragma unroll
  for (int kc = 0; kc < 4; ++kc) {
    int cidx = 4 * kc + cb;
    Ar[kc].x = umat[col * 16 + cidx];
    Ar[kc].y = umat[col * 16 + cidx + 1];
    Ai[kc].x = umat[256 + col * 16 + cidx];
    Ai[kc].y = umat[256 + col * 16 + cidx + 1];
  }

  float acc0 = 0.f, acc1 = 0.f, acc2 = 0.f, acc3 = 0.f;
  const float PIH = 1.57079632679489662f;

#pragma unroll
  for (int ch = 0; ch < 3; ++ch) {
    const float* base = x + ((size_t)ch * HW + (size_t)pi) * HW + pj;
    float x00 = base[0], x01 = base[1], x10 = base[HW], x11 = base[HW + 1];
    float s0, c0, s1, c1, s2, c2, s3, c3;
    __sincosf(PIH * x00, &s0, &c0);
    __sincosf(PIH * x01, &s1, &c1);
    __sincosf(PIH * x10, &s2, &c2);
    __sincosf(PIH * x11, &s3, &c3);

    // psi[a] = t0[a>>2] * t1[a&3], a = b0*8+b1*4+b2*2+b3.
    // B chunk kc, vgpr v holds row a = 4*kc + v + 2*half -> a>>2 == kc,
    // a&3 == v + 2*half. So B[kc][v] = t0[kc] * t1sel[v].
    float t00 = c0 * c1, t01 = c0 * s1, t02 = s0 * c1, t03 = s0 * s1;
    float u0 = half ? (s2 * c3) : (c2 * c3);   // t1[2*half + 0]
    float u1 = half ? (s2 * s3) : (c2 * s3);   // t1[2*half + 1]

    v2f B[4];
    B[0].x = t00 * u0; B[0].y = t00 * u1;
    B[1].x = t01 * u0; B[1].y = t01 * u1;
    B[2].x = t02 * u0; B[2].y = t02 * u1;
    B[3].x = t03 * u0; B[3].y = t03 * u1;

    v8f yr = {}, yi = {};
#pragma unroll
    for (int kc = 0; kc < 4; ++kc)
      yr = __builtin_amdgcn_wmma_f32_16x16x4_f32(false, Ar[kc], false, B[kc],
                                                 (short)0, yr, false, false);
#pragma unroll
    for (int kc = 0; kc < 4; ++kc)
      yi = __builtin_amdgcn_wmma_f32_16x16x4_f32(false, Ai[kc], false, B[kc],
                                                 (short)0, yi, false, false);

    // D layout: this lane holds Y[M = 8*half + r][N = col], r = 0..7
    float S = 0.f, T1 = 0.f, T2 = 0.f, T3 = 0.f;
#pragma unroll
    for (int r = 0; r < 8; ++r) {
      float p = yr[r] * yr[r] + yi[r] * yi[r];
      S  += p;
      T1 += ((r >> 2) & 1) ? -p : p;   // sign from bit b1
      T2 += ((r >> 1) & 1) ? -p : p;   // sign from bit b2
      T3 += ((r >> 0) & 1) ? -p : p;   // sign from bit b3
    }
    // combine the two M-halves (bit b0) across lane pairs (lane ^ 16)
    float Sp  = swap16(S);
    float T1p = swap16(T1);
    float T2p = swap16(T2);
    float T3p = swap16(T3);
    acc0 += half ? (Sp - S) : (S - Sp);
    acc1 += T1 + T1p;
    acc2 += T2 + T2p;
    acc3 += T3 + T3p;
  }

  if (valid && half == 0) {
    // out flat = q_out[i][j][k] at (i*511 + j)*4 + k  == reference reshape
    float4 v = make_float4(acc0, acc1, acc2, acc3);
    *reinterpret_cast<float4*>(out + (size_t)pix * 4) = v;
  }
}

// ---------------------------------------------------------------------------
// Host: replicate numpy.random.RandomState(42) gate-structure sampling
// (MT19937 + legacy masked-rejection bounded draws + Fisher-Yates shuffle).
// ---------------------------------------------------------------------------
namespace {

struct MT { uint32_t mt[624]; int idx; };

static void mt_seed(MT& s, uint32_t seed) {
  s.mt[0] = seed;
  for (int i = 1; i < 624; ++i)
    s.mt[i] = 1812433253u * (s.mt[i - 1] ^ (s.mt[i - 1] >> 30)) + (uint32_t)i;
  s.idx = 624;
}
static uint32_t mt_next(MT& s) {
  if (s.idx >= 624) {
    for (int i = 0; i < 624; ++i) {
      uint32_t y = (s.mt[i] & 0x80000000u) | (s.mt[(i + 1) % 624] & 0x7fffffffu);
      uint32_t v = s.mt[(i + 397) % 624] ^ (y >> 1);
      if (y & 1u) v ^= 2567483615u;
      s.mt[i] = v;
    }
    s.idx = 0;
  }
  uint32_t y = s.mt[s.idx++];
  y ^= y >> 11; y ^= (y << 7) & 2636928640u; y ^= (y << 15) & 4022730752u; y ^= y >> 18;
  return y;
}
static double mt_double(MT& s) {           // numpy random_sample()
  uint32_t a = mt_next(s) >> 5, b = mt_next(s) >> 6;
  return (a * 67108864.0 + b) / 9007199254740992.0;
}
static uint32_t mt_interval(MT& s, uint32_t mx) {  // legacy rk_interval: [0, mx]
  if (mx == 0) return 0;
  uint32_t mask = mx;
  mask |= mask >> 1; mask |= mask >> 2; mask |= mask >> 4;
  mask |= mask >> 8; mask |= mask >> 16;
  uint32_t v;
  do { v = mt_next(s) & mask; } while (v > mx);
  return v;
}

static OpList build_ops() {
  OpList ol; ol.n = 0;
  MT rng; mt_seed(rng, 42u);
  for (int l = 0; l < 2; ++l) {
    int i = 0, guard = 0;
    while (i < 4 && ol.n < 63 && guard++ < 100000) {
      if (mt_double(rng) > 0.3) {                       // rotation
        int g    = (int)mt_interval(rng, 2);            // randint(3)
        int wire = (int)mt_interval(rng, 3);            // randint(4)
        ol.ops[ol.n++] = (unsigned)(g | (wire << 2) | (l << 6) | (i << 7));
        ++i;
      } else {                                          // CNOT: choice(4,2,replace=False)
        int arr[4] = {0, 1, 2, 3};                      // permutation via shuffle
        for (int k = 3; k >= 1; --k) {
          int j = (int)mt_interval(rng, (uint32_t)k);
          int t = arr[k]; arr[k] = arr[j]; arr[j] = t;
        }
        ol.ops[ol.n++] = (unsigned)(3 | (arr[0] << 2) | (arr[1] << 4) | (l << 6));
      }
    }
  }
  return ol;
}

} // namespace

extern "C" void kernel_launch(void* const* d_in, const int* in_sizes, int n_in,
                              void* d_out, int out_size, void* d_ws, size_t ws_size,
                              hipStream_t stream)
{
  const float* x   = (const float*)d_in[0];   // [1,3,512,512] f32
  const float* wgt = (const float*)d_in[1];   // [2,4] f32
  float* out  = (float*)d_out;                // [1,4,511,511] f32
  float* umat = (float*)d_ws;                 // 512 floats: Ur | Ui

  OpList ol = build_ops();                    // deterministic (seed 42)
  quanv_build_u<<<1, 32, 0, stream>>>(wgt, umat, ol);

  const int pix_per_block = 128;              // 8 waves x 16 pixels
  const int blocks = (NPIX + pix_per_block - 1) / pix_per_block;
  quanv_main<<<blocks, 256, 0, stream>>>(x, umat, out);
}